// TensorDlt_48215302865189
// MI455X (gfx1250) — compile-verified
//
#include <hip/hip_runtime.h>
#include <math.h>

// MI455X (gfx1250) batched 8x8 DLT pinv-solve.
// Memory-bound (~25 MB @ 23.3 TB/s ~ 1.1us). Strategy:
//  - pack 2 systems block-diagonally into a 16x16 tile per wave32
//  - Newton-Schulz pseudoinverse in f32 via chained V_WMMA_F32_16X16X4_F32
//    (A is exactly representable in f32: trunc'd integers < 2^24)
//  - f64 iterative refinement (O(n^2)) recovers double accuracy
// The NS loop is wave-private (per-wave in-order LDS => no block barriers),
// and the loop-invariant A fragments are held in VGPRs across all iterations.

typedef __attribute__((ext_vector_type(2))) float v2f;
typedef __attribute__((ext_vector_type(8))) float v8f;

#define BLOCK 256
#define WAVES 8            // wave32 => 8 waves per block
#define SYS_PER_BLOCK 16   // 2 systems per wave (block-diag 16x16 tile)
#define NS_ITERS 24        // Newton-Schulz iterations (quadratic convergence)
#define REFINE_ITERS 3     // f64 iterative refinement rounds

struct WaveTiles {
  float  A[16][17];        // block-diag A tile, f32 (exact), padded stride 17
  float  X[16][17];        // running pinv estimate
  float  S[16][17];        // 2I - A*X scratch
  double A64[2][8][8];     // exact f64 A per system (for refinement residual)
  double b[16];            // rhs (2 systems x 8), f64
  double h[16];            // solution
  double r[16];            // residual
};

__global__ __launch_bounds__(BLOCK)
void dlt_pinv_kernel(const float* __restrict__ x,
                     const float* __restrict__ xp,
                     float* __restrict__ out,
                     int nsys) {
  __shared__ WaveTiles wt[WAVES];
  __shared__ float rsum[SYS_PER_BLOCK][8];
  __shared__ float csum[SYS_PER_BLOCK][8];
  __shared__ float scaleArr[SYS_PER_BLOCK];

  const int tid  = threadIdx.x;
  const int wave = tid >> 5;
  const int lane = tid & 31;
  const int s0   = blockIdx.x * SYS_PER_BLOCK;

  // ---- Phase 0: zero the A tile (off-diagonal 8x8 blocks must be 0) ----
  {
    float* a = &wt[wave].A[0][0];
    for (int i = lane; i < 16 * 17; i += 32) a[i] = 0.0f;
  }
  __syncthreads();

  // ---- Phase 1: build A (trunc'd, faithful to reference), b ----
  // one thread per (system, point): 16 systems * 4 points = 64 threads
  if (tid < 64) {
    const int sysl = tid >> 2;       // 0..15 local system
    const int p    = tid & 3;        // point 0..3
    const int g    = s0 + sysl;
    double u = 0.0, v = 0.0, up = 0.0, vp = 0.0;
    if (g < nsys) {
      const float2 pa = *(const float2*)(x  + g * 8 + p * 2);
      const float2 pb = *(const float2*)(xp + g * 8 + p * 2);
      u  = (double)pa.x;  v  = (double)pa.y;
      up = (double)pb.x;  vp = (double)pb.y;
    }
    double ax[8] = {0.0, 0.0, 0.0, trunc(-u), trunc(-v), -1.0,
                    trunc(vp * u), trunc(vp * v)};
    double ay[8] = {trunc(u), trunc(v), 1.0, 0.0, 0.0, 0.0,
                    trunc(-up * u), trunc(-up * v)};
    const int w  = sysl >> 1;        // owning wave
    const int si = sysl & 1;         // slot within wave (0/1)
    const int rb = si * 8;           // block-diag row/col base
    const int r0 = rb + 2 * p;       // interleaved rows: even=x-row, odd=y-row
#pragma unroll
    for (int j = 0; j < 8; ++j) {
      wt[w].A64[si][2 * p + 0][j] = ax[j];
      wt[w].A64[si][2 * p + 1][j] = ay[j];
      wt[w].A[r0 + 0][rb + j] = (float)ax[j];
      wt[w].A[r0 + 1][rb + j] = (float)ay[j];
    }
    wt[w].b[r0 + 0] = -vp;
    wt[w].b[r0 + 1] = up;
  }
  __syncthreads();

  // ---- Phase 2: per-system scale 1/(||A||_1 * ||A||_inf) for NS init ----
  {
    const int sys = tid >> 4;        // 0..15
    const int q   = tid & 15;
    const int w = sys >> 1, si = sys & 1, bse = si * 8;
    float s = 0.0f;
    if (q < 8) {                     // row sum (inf-norm candidate)
#pragma unroll
      for (int j = 0; j < 8; ++j) s += fabsf(wt[w].A[bse + q][bse + j]);
      rsum[sys][q] = s;
    } else {                         // column sum (1-norm candidate)
      const int c = q - 8;
#pragma unroll
      for (int i = 0; i < 8; ++i) s += fabsf(wt[w].A[bse + i][bse + c]);
      csum[sys][c] = s;
    }
  }
  __syncthreads();
  if (tid < SYS_PER_BLOCK) {
    float mr = 0.0f, mc = 0.0f;
#pragma unroll
    for (int j = 0; j < 8; ++j) {
      mr = fmaxf(mr, rsum[tid][j]);
      mc = fmaxf(mc, csum[tid][j]);
    }
    const float prod = mr * mc;
    scaleArr[tid] = (prod > 0.0f) ? (1.0f / prod) : 0.0f;
  }
  __syncthreads();

  // ---- Phase 3: X0 = scale * A^T (block-diag transpose) ----
  for (int idx = tid; idx < WAVES * 256; idx += BLOCK) {
    const int w = idx >> 8, rem = idx & 255, i = rem >> 4, j = rem & 15;
    wt[w].X[i][j] = wt[w].A[j][i] * scaleArr[w * 2 + (i >> 3)];
  }
  __syncthreads();   // last cross-wave hand-off before wave-private region

  // ---- Phase 4: Newton-Schulz  X <- X (2I - A X)  via WMMA f32 ----
  // Wave-private: per-wave in-order LDS (DScnt) gives cross-lane visibility
  // within the wave; no s_barrier needed here.
  // ISA fragment layouts (wave32):
  //  A 16x4: lanes 0-15 -> (M=lane, K=0/1), lanes 16-31 -> (M=lane-16, K=2/3)
  //  B 4x16: v0: (K=0/2, N=lane&15) ; v1: (K=1/3, N=lane&15)
  //  C/D 16x16: VGPR r: lanes 0-15 -> (M=r, N), lanes 16-31 -> (M=r+8, N)
  const int hf   = lane >> 4;        // half-wave select
  const int mrow = lane & 15;        // A-fragment row
  const int ncol = lane & 15;        // B/C-fragment column

  // A is loop-invariant: hold its 4 k-step fragments in VGPRs for all
  // NS_ITERS iterations (removes 1/4 of the in-loop LDS loads).
  v2f afrag[4];
#pragma unroll
  for (int s = 0; s < 4; ++s) {
    const int k0 = 4 * s + hf * 2;
    afrag[s].x = wt[wave].A[mrow][k0];
    afrag[s].y = wt[wave].A[mrow][k0 + 1];
  }

  for (int it = 0; it < NS_ITERS; ++it) {
    // T = A * X   (16x16x16 = 4 chained 16x16x4 WMMAs, A frags in registers)
    v8f acc = {0.f, 0.f, 0.f, 0.f, 0.f, 0.f, 0.f, 0.f};
#pragma unroll
    for (int s = 0; s < 4; ++s) {
      const int k0 = 4 * s + hf * 2;
      v2f bf;
      bf.x = wt[wave].X[k0][ncol];
      bf.y = wt[wave].X[k0 + 1][ncol];
      acc = __builtin_amdgcn_wmma_f32_16x16x4_f32(
          false, afrag[s], false, bf, (short)0, acc, false, false);
    }
    // S = 2I - T
#pragma unroll
    for (int rr = 0; rr < 8; ++rr) {
      const int mi = rr + hf * 8;
      wt[wave].S[mi][ncol] = ((mi == ncol) ? 2.0f : 0.0f) - acc[rr];
    }
    __builtin_amdgcn_wave_barrier();   // scheduling fence: S stores <-> S loads
    // Xn = X * S
    v8f acc2 = {0.f, 0.f, 0.f, 0.f, 0.f, 0.f, 0.f, 0.f};
#pragma unroll
    for (int s = 0; s < 4; ++s) {
      const int k0 = 4 * s + hf * 2;
      v2f af, bf;
      af.x = wt[wave].X[mrow][k0];
      af.y = wt[wave].X[mrow][k0 + 1];
      bf.x = wt[wave].S[k0][ncol];
      bf.y = wt[wave].S[k0 + 1][ncol];
      acc2 = __builtin_amdgcn_wmma_f32_16x16x4_f32(
          false, af, false, bf, (short)0, acc2, false, false);
    }
    __builtin_amdgcn_wave_barrier();   // X reads above <-> X overwrite below
#pragma unroll
    for (int rr = 0; rr < 8; ++rr)
      wt[wave].X[rr + hf * 8][ncol] = acc2[rr];
    __builtin_amdgcn_wave_barrier();
  }

  // ---- Phase 5: h = X b, then f64 refinement h += X (b - A h) ----
  // Still wave-private (lanes 0-15 of each wave; per-wave LDS ordering).
  if (lane < 16) {
    const int i  = lane;             // tile row
    const int si = i >> 3;
    const int jb = si * 8;
    double hv = 0.0;
#pragma unroll
    for (int j = 0; j < 8; ++j)
      hv += (double)wt[wave].X[i][jb + j] * wt[wave].b[jb + j];
    wt[wave].h[i] = hv;
  }
  __builtin_amdgcn_wave_barrier();
  for (int t = 0; t < REFINE_ITERS; ++t) {
    if (lane < 16) {
      const int i = lane, si = i >> 3, rr = i & 7, jb = si * 8;
      double res = wt[wave].b[i];
#pragma unroll
      for (int j = 0; j < 8; ++j)
        res -= wt[wave].A64[si][rr][j] * wt[wave].h[jb + j];
      wt[wave].r[i] = res;
    }
    __builtin_amdgcn_wave_barrier();
    if (lane < 16) {
      const int i = lane, si = i >> 3, jb = si * 8;
      double c = 0.0;
#pragma unroll
      for (int j = 0; j < 8; ++j)
        c += (double)wt[wave].X[i][jb + j] * wt[wave].r[jb + j];
      wt[wave].h[i] += c;
    }
    __builtin_amdgcn_wave_barrier();
  }

  __syncthreads();   // store phase reads other waves' h tiles

  // ---- Phase 6: store H = [h0..h7, 1] per system (coalesced) ----
  for (int idx = tid; idx < SYS_PER_BLOCK * 9; idx += BLOCK) {
    const int sysl = idx / 9, k = idx % 9;
    const int g = s0 + sysl;
    if (g < nsys) {
      const int w = sysl >> 1;
      const int i = (sysl & 1) * 8 + k;
      out[g * 9 + k] = (k == 8) ? 1.0f : (float)wt[w].h[i];
    }
  }
}

extern "C" void kernel_launch(void* const* d_in, const int* in_sizes, int n_in,
                              void* d_out, int out_size, void* d_ws, size_t ws_size,
                              hipStream_t stream) {
  const float* x  = (const float*)d_in[0];
  const float* xp = (const float*)d_in[1];
  float* out = (float*)d_out;
  const int nsys = in_sizes[0] / 8;                       // 262144
  const int blocks = (nsys + SYS_PER_BLOCK - 1) / SYS_PER_BLOCK;  // 16384
  dlt_pinv_kernel<<<blocks, BLOCK, 0, stream>>>(x, xp, out, nsys);
}